// MHAWrapper_18047452578729
// MI455X (gfx1250) — compile-verified
//
#include <hip/hip_runtime.h>

#define DMODEL 1280
#define NHEAD  20
#define DHEAD  64
#define SCACHE 8192
#define STOT   (SCACHE + 1)
#define CS     512
#define NCHUNK ((STOT + CS - 1) / CS)   // 17

typedef __attribute__((ext_vector_type(2))) float v2f;
typedef __attribute__((ext_vector_type(8))) float v8f;

// y[row0 + 0..15] = bias[row0+..] + sum_k W[row0+m][k] * x[k]
// One wave, V_WMMA_F32_16X16X4_F32.
// A 16x4 layout: lanes 0-15 hold M=lane with K0/K1 in VGPR0/1; lanes 16-31 K2/K3.
// B 4x16: every lane supplies the same x slice, so ALL columns of B are copies
// of x[k..k+3]; columns of D are independent, and column 0 (lanes 0 and 16)
// is exactly bias + W.x.  This keeps the inner loop branch-free (no EXEC
// masking on the x loads).
__device__ __forceinline__ void gemv16_wmma(const float* __restrict__ W, int ldw, int row0,
                                            const float* __restrict__ x, int K,
                                            const float* __restrict__ bias,
                                            float* __restrict__ y) {
  const int lane = threadIdx.x & 31;
  const int half = lane >> 4;   // 0 or 1
  const int m    = lane & 15;   // A-matrix row
  v8f c = {};
  if (bias && m == 0) {
    #pragma unroll
    for (int e = 0; e < 8; ++e) c[e] = bias[row0 + half * 8 + e];
  }
  const float* wrow = W + (size_t)(row0 + m) * ldw + 2 * half;
  const float* xp   = x + 2 * half;
  #pragma unroll 8
  for (int k = 0; k < K; k += 4) {
    v2f a, b;
    a.x = wrow[k];
    a.y = wrow[k + 1];
    b.x = xp[k];        // uniform broadcast load, no masking
    b.y = xp[k + 1];
    c = __builtin_amdgcn_wmma_f32_16x16x4_f32(false, a, false, b, (short)0, c,
                                              false, false);
  }
  if (m == 0) {
    #pragma unroll
    for (int e = 0; e < 8; ++e) y[row0 + half * 8 + e] = c[e];
  }
}

// qkv = x @ [Wq; Wk; Wv]^T (+bq / - / +bv).  grid = 240 waves of 32.
__global__ void __launch_bounds__(32)
qkv_proj_kernel(const float* __restrict__ x,
                const float* __restrict__ Wq, const float* __restrict__ bq,
                const float* __restrict__ Wk,
                const float* __restrict__ Wv, const float* __restrict__ bv,
                float* __restrict__ qkv) {
  const int r0 = blockIdx.x * 16;
  if (r0 < DMODEL) {
    gemv16_wmma(Wq, DMODEL, r0, x, DMODEL, bq, qkv);
  } else if (r0 < 2 * DMODEL) {
    gemv16_wmma(Wk, DMODEL, r0 - DMODEL, x, DMODEL, nullptr, qkv + DMODEL);
  } else {
    gemv16_wmma(Wv, DMODEL, r0 - 2 * DMODEL, x, DMODEL, bv, qkv + 2 * DMODEL);
  }
}

// One block per (chunk, head): streams its 512x64 tile of k/v cache exactly
// once, copies it to k_out/v_out, and produces flash-attention partials
// (chunk max, exp-sum, unnormalized weighted-V).
__global__ void __launch_bounds__(256)
attn_pass_kernel(const float* __restrict__ k_cache, const float* __restrict__ v_cache,
                 const float* __restrict__ qkv,
                 float* __restrict__ k_out, float* __restrict__ v_out,
                 float* __restrict__ part_m, float* __restrict__ part_l,
                 float* __restrict__ part_wv) {
  const int chunk = blockIdx.x;
  const int h     = blockIdx.y;
  const int tid   = threadIdx.x;
  const int lane  = tid & 31;
  const int wave  = tid >> 5;
  const int col0  = h * DHEAD;
  const int j0    = chunk * CS;
  const int jend  = (j0 + CS < STOT) ? (j0 + CS) : STOT;
  const int nrows = jend - j0;

  __shared__ float q_s[DHEAD];
  __shared__ float sc[CS];
  __shared__ float red[8];
  __shared__ float wv_s[4][DHEAD];

  // scale = Dh^-0.25 applied to both q and k  ==>  fold Dh^-0.5 into q
  if (tid < DHEAD) q_s[tid] = qkv[col0 + tid] * 0.125f;
  __syncthreads();

  // Pass 1: scores (one row per wave) + copy K tile to output
  for (int j = j0 + wave; j < jend; j += 8) {
    const float* krow = (j < SCACHE) ? (k_cache + (size_t)j * DMODEL + col0)
                                     : (qkv + DMODEL + col0);  // appended k_new
    if (j + 8 < SCACHE)  // stay ahead of the 84MB stream
      __builtin_prefetch(k_cache + (size_t)(j + 8) * DMODEL + col0 + 2 * lane, 0, 1);
    float2 kk = *(const float2*)(krow + 2 * lane);
    *((float2*)(k_out + (size_t)j * DMODEL + col0) + lane) = kk;
    float p = kk.x * q_s[2 * lane] + kk.y * q_s[2 * lane + 1];
    #pragma unroll
    for (int off = 16; off > 0; off >>= 1) p += __shfl_down(p, off);
    if (lane == 0) sc[j - j0] = p;
  }
  __syncthreads();

  // Chunk max
  float mloc = -3.40282347e38f;
  for (int i = tid; i < nrows; i += 256) mloc = fmaxf(mloc, sc[i]);
  #pragma unroll
  for (int off = 16; off > 0; off >>= 1) mloc = fmaxf(mloc, __shfl_down(mloc, off));
  if (lane == 0) red[wave] = mloc;
  __syncthreads();
  float M = red[0];
  #pragma unroll
  for (int w = 1; w < 8; ++w) M = fmaxf(M, red[w]);

  // exp and sum
  float lloc = 0.0f;
  for (int i = tid; i < nrows; i += 256) {
    float e = __expf(sc[i] - M);
    sc[i] = e;
    lloc += e;
  }
  __syncthreads();
  #pragma unroll
  for (int off = 16; off > 0; off >>= 1) lloc += __shfl_down(lloc, off);
  if (lane == 0) red[wave] = lloc;
  __syncthreads();
  float L = 0.0f;
  #pragma unroll
  for (int w = 0; w < 8; ++w) L += red[w];

  // Pass 2: weighted-V partial + copy V tile to output
  const int g = tid >> 6;   // 4 row-groups
  const int d = tid & 63;   // head dim
  float acc = 0.0f;
  for (int j = j0 + g; j < jend; j += 4) {
    const float* vrow = (j < SCACHE) ? (v_cache + (size_t)j * DMODEL + col0)
                                     : (qkv + 2 * DMODEL + col0);  // v_new
    if (j + 4 < SCACHE)
      __builtin_prefetch(v_cache + (size_t)(j + 4) * DMODEL + col0 + d, 0, 1);
    float vv = vrow[d];
    v_out[(size_t)j * DMODEL + col0 + d] = vv;
    acc += sc[j - j0] * vv;
  }
  wv_s[g][d] = acc;
  __syncthreads();
  if (g == 0) {
    part_wv[(size_t)(h * NCHUNK + chunk) * DHEAD + d] =
        wv_s[0][d] + wv_s[1][d] + wv_s[2][d] + wv_s[3][d];
  }
  if (tid == 0) {
    part_m[h * NCHUNK + chunk] = M;
    part_l[h * NCHUNK + chunk] = L;
  }
}

// Merge chunk partials per head (log-sum-exp combine), normalize.
__global__ void __launch_bounds__(64)
combine_kernel(const float* __restrict__ part_m, const float* __restrict__ part_l,
               const float* __restrict__ part_wv, float* __restrict__ wv) {
  const int h = blockIdx.x;
  const int d = threadIdx.x;
  float M = -3.40282347e38f;
  for (int i = 0; i < NCHUNK; ++i) M = fmaxf(M, part_m[h * NCHUNK + i]);
  float L = 0.0f, acc = 0.0f;
  for (int i = 0; i < NCHUNK; ++i) {
    float w = __expf(part_m[h * NCHUNK + i] - M);
    L   += w * part_l[h * NCHUNK + i];
    acc += w * part_wv[(size_t)(h * NCHUNK + i) * DHEAD + d];
  }
  wv[h * DHEAD + d] = acc / L;
}

// out = wv @ Wo^T + bo.  grid = 80 waves of 32.
__global__ void __launch_bounds__(32)
out_proj_kernel(const float* __restrict__ wv, const float* __restrict__ Wo,
                const float* __restrict__ bo, float* __restrict__ out) {
  gemv16_wmma(Wo, DMODEL, blockIdx.x * 16, wv, DMODEL, bo, out);
}

extern "C" void kernel_launch(void* const* d_in, const int* in_sizes, int n_in,
                              void* d_out, int out_size, void* d_ws, size_t ws_size,
                              hipStream_t stream) {
  const float* x  = (const float*)d_in[0];
  const float* kc = (const float*)d_in[1];
  const float* vc = (const float*)d_in[2];
  const float* Wq = (const float*)d_in[3];
  const float* bq = (const float*)d_in[4];
  const float* Wk = (const float*)d_in[5];
  const float* Wv = (const float*)d_in[6];
  const float* bv = (const float*)d_in[7];
  const float* Wo = (const float*)d_in[8];
  const float* bo = (const float*)d_in[9];
  (void)in_sizes; (void)n_in; (void)out_size; (void)ws_size;

  // Output tuple (out, k, v) flattened in return order.
  float* out   = (float*)d_out;
  float* k_out = out + DMODEL;
  float* v_out = k_out + (size_t)STOT * DMODEL;

  // Workspace layout (floats).
  float* ws      = (float*)d_ws;
  float* qkv     = ws;                                   // 3*1280
  float* part_m  = qkv + 3 * DMODEL;                     // 20*17
  float* part_l  = part_m + NHEAD * NCHUNK;              // 20*17
  float* part_wv = part_l + NHEAD * NCHUNK;              // 20*17*64
  float* wv      = part_wv + NHEAD * NCHUNK * DHEAD;     // 1280

  qkv_proj_kernel<<<dim3(3 * DMODEL / 16), dim3(32), 0, stream>>>(
      x, Wq, bq, Wk, Wv, bv, qkv);

  attn_pass_kernel<<<dim3(NCHUNK, NHEAD), dim3(256), 0, stream>>>(
      kc, vc, qkv, k_out, v_out, part_m, part_l, part_wv);

  combine_kernel<<<dim3(NHEAD), dim3(DHEAD), 0, stream>>>(
      part_m, part_l, part_wv, wv);

  out_proj_kernel<<<dim3(DMODEL / 16), dim3(32), 0, stream>>>(wv, Wo, bo, out);
}